// blocks_mse_47665547051143
// MI455X (gfx1250) — compile-verified
//
#include <hip/hip_runtime.h>
#include <hip/hip_bf16.h>

typedef float v2f __attribute__((ext_vector_type(2)));
typedef float v8f __attribute__((ext_vector_type(8)));

#define BB 64
#define CC 768
#define NN 576          // 24*24 positions
#define KB 3            // blocks
#define CHUNK 192       // NN / KB
#define VDIM (CC * KB)  // 2304 concatenated channels
#define THREADS 512     // 16 waves of 32

// Kernel 1: one workgroup per (map, batch) unit.
//   Phase 1: heatmap h[n] = sum_c X[c,n]      (HBM pass, coalesced)
//   Phase 2: rank each position (stable desc) -> chunk id -> selection matrix in LDS
//   Phase 3: pooled[C,3] = X[C,576] x S[576,3] via V_WMMA_F32_16X16X4_F32 (L2 re-read)
__global__ __launch_bounds__(THREADS)
void pool_kernel(const float* __restrict__ x1, const float* __restrict__ x2,
                 float* __restrict__ vout /* [2][64][VDIM] */)
{
    __shared__ float h[NN];
    __shared__ float selv[NN * 16];   // 36 KB selection matrix (cols 3..15 zero)

    const int unit = blockIdx.x;      // 0..127
    const int map  = unit >> 6;
    const int b    = unit & 63;
    const float* __restrict__ X = (map == 0 ? x1 : x2) + (size_t)b * CC * NN;
    const int t = threadIdx.x;

    // ---- Phase 1: per-position channel sum (rank-equivalent to the mean) ----
    for (int n = t; n < NN; n += THREADS) {
        float s0 = 0.f, s1 = 0.f, s2 = 0.f, s3 = 0.f;
        const float* p = X + n;
        for (int c = 0; c < CC; c += 4) {
            s0 += p[(c + 0) * NN];
            s1 += p[(c + 1) * NN];
            s2 += p[(c + 2) * NN];
            s3 += p[(c + 3) * NN];
        }
        h[n] = (s0 + s1) + (s2 + s3);
    }
    __syncthreads();

    // ---- Phase 2: stable descending rank -> chunk -> selection matrix ----
    for (int n = t; n < NN; n += THREADS) {
        const float hn = h[n];
        int r = 0;
        for (int m = 0; m < NN; ++m) {
            const float hm = h[m];
            r += (hm > hn) || (hm == hn && m < n);
        }
        const int k = r / CHUNK;   // 0..2
        #pragma unroll
        for (int kk = 0; kk < 16; ++kk)
            selv[n * 16 + kk] = (kk == k) ? (1.0f / (float)CHUNK) : 0.0f;
    }
    __syncthreads();

    // ---- Phase 3: segmented mean as fp32 WMMA GEMM ----
    // A (16x4 f32): lane-half hh, VGPR v -> K = v + 2*hh, M = lane%16 (channel)
    // B (4x16 f32): lane-half hh, VGPR v -> K = v + 2*hh, N = lane%16 (chunk slot)
    // D (16x16 f32): VGPR r -> M = r + 8*hh, N = lane%16
    const int wave = t >> 5;          // 0..15
    const int lane = t & 31;
    const int hh   = lane >> 4;       // 0/1
    const int lm   = lane & 15;
    float* vb = vout + (size_t)unit * VDIM;

    for (int i = 0; i < 3; ++i) {     // 16 waves x 3 tiles = 48 channel tiles
        const int c0 = (wave * 3 + i) * 16;
        v8f acc = {};
        const float* pa = X + (size_t)(c0 + lm) * NN + 2 * hh;     // L2-resident
        const float* ps = &selv[(2 * hh) * 16 + lm];
        for (int j = 0; j < NN / 4; ++j) {                         // 144 K-steps
            v2f a = *(const v2f*)(pa + 4 * j);                     // 8B aligned
            v2f bm;
            bm.x = ps[(4 * j + 0) * 16];
            bm.y = ps[(4 * j + 1) * 16];
            acc = __builtin_amdgcn_wmma_f32_16x16x4_f32(
                      false, a, false, bm, (short)0, acc, false, false);
        }
        if (lm < KB) {                 // only chunk slots 0..2 are real
            #pragma unroll
            for (int r = 0; r < 8; ++r) {
                const int c = c0 + r + 8 * hh;
                vb[lm * CC + c] = acc[r];   // concat layout: j = k*C + c
            }
        }
    }
}

// Kernel 2: per-batch L2 norms + squared-diff partial sum (deterministic tree reduce)
__global__ __launch_bounds__(256)
void normdiff_kernel(const float* __restrict__ vout, float* __restrict__ partial)
{
    __shared__ float red[256];
    const int b = blockIdx.x;
    const int t = threadIdx.x;
    const float* v1 = vout + (size_t)b * VDIM;
    const float* v2 = vout + (size_t)(BB + b) * VDIM;

    float s1 = 0.f, s2 = 0.f;
    for (int j = t; j < VDIM; j += 256) {
        const float a = v1[j], c = v2[j];
        s1 += a * a; s2 += c * c;
    }
    red[t] = s1; __syncthreads();
    for (int o = 128; o > 0; o >>= 1) { if (t < o) red[t] += red[t + o]; __syncthreads(); }
    float n1 = red[0]; __syncthreads();
    red[t] = s2; __syncthreads();
    for (int o = 128; o > 0; o >>= 1) { if (t < o) red[t] += red[t + o]; __syncthreads(); }
    float n2 = red[0]; __syncthreads();

    n1 = fmaxf(sqrtf(n1), 1e-12f);
    n2 = fmaxf(sqrtf(n2), 1e-12f);

    float d = 0.f;
    for (int j = t; j < VDIM; j += 256) {
        const float e = v1[j] / n1 - v2[j] / n2;
        d += e * e;
    }
    red[t] = d; __syncthreads();
    for (int o = 128; o > 0; o >>= 1) { if (t < o) red[t] += red[t + o]; __syncthreads(); }
    if (t == 0) partial[b] = red[0];
}

// Kernel 3: final scalar
__global__ void final_kernel(const float* __restrict__ partial, float* __restrict__ out)
{
    __shared__ float red[BB];
    const int t = threadIdx.x;
    red[t] = partial[t];
    __syncthreads();
    for (int o = 32; o > 0; o >>= 1) { if (t < o) red[t] += red[t + o]; __syncthreads(); }
    if (t == 0) out[0] = red[0] / (float)((size_t)BB * VDIM);
}

extern "C" void kernel_launch(void* const* d_in, const int* in_sizes, int n_in,
                              void* d_out, int out_size, void* d_ws, size_t ws_size,
                              hipStream_t stream)
{
    (void)in_sizes; (void)n_in; (void)out_size; (void)ws_size;
    const float* x1 = (const float*)d_in[0];
    const float* x2 = (const float*)d_in[1];
    // d_in[2]=logit_scale, d_in[3]=weights are unused by the reference output.
    float* vout    = (float*)d_ws;              // 2*64*2304 floats = 1.18 MB
    float* partial = vout + 2 * BB * VDIM;      // +64 floats

    pool_kernel<<<2 * BB, THREADS, 0, stream>>>(x1, x2, vout);
    normdiff_kernel<<<BB, 256, 0, stream>>>(vout, partial);
    final_kernel<<<1, BB, 0, stream>>>(partial, (float*)d_out);
}